// UberNet_40381282517638
// MI455X (gfx1250) — compile-verified
//
#include <hip/hip_runtime.h>
#include <math.h>

#define B_ 64
#define T_ 1000
#define H_ 64
#define G_ 192              // 3*H
#define D_ 200
#define ROWS_PER_SIDE (B_*T_)        // 64000
#define ROWS_TOTAL    (2*ROWS_PER_SIDE)

typedef __attribute__((ext_vector_type(2))) float v2f;
typedef __attribute__((ext_vector_type(8))) float v8f;

// D(16x16 f32) = A(16x4 f32) * B(4x16 f32) + C  -- full-precision CDNA5 WMMA
__device__ __forceinline__ v8f wmma_f32(v2f a, v2f b, v8f c) {
    return __builtin_amdgcn_wmma_f32_16x16x4_f32(
        /*neg_a=*/false, a, /*neg_b=*/false, b,
        /*c_mod=*/(short)0, c, /*reuse_a=*/false, /*reuse_b=*/false);
}

__device__ __forceinline__ float sigmoid_(float x) { return 1.0f / (1.0f + __expf(-x)); }

// ---------------------------------------------------------------------------
// Row enumeration everywhere: row = side*64000 + t*64 + b  (t-major).
// xp layout: [side][t][b][192]   h layout: [side][t][b][64]
// -> per-timestep slabs are contiguous (48 KB / 16 KB).
// ---------------------------------------------------------------------------

// Kernel 1: fused embedding gather + input projection, layer 0.
__global__ void k_xp0(const int* __restrict__ x, const float* __restrict__ emb,
                      const float* __restrict__ WihC, const float* __restrict__ bihC,
                      const float* __restrict__ WihE, const float* __restrict__ bihE,
                      float* __restrict__ xp)
{
    const int tid  = threadIdx.x;
    const int lane = tid & 31, wave = tid >> 5;
    const int gw   = blockIdx.x * 8 + wave;            // global wave id
    const int mtile = gw / 12, ntile = gw % 12;        // 8000 M-tiles, 12 N-tiles
    const int lmod = lane & 15;
    const int kof  = (lane >> 4) * 2;                  // K sub-offset per half-wave
    const int side = (mtile * 16) / ROWS_PER_SIDE;     // uniform within tile
    const float* Wih = side ? WihE : WihC;
    const float* bih = side ? bihE : bihC;

    // per-lane A row (t-major) -> token -> embedding row
    const int row = mtile * 16 + lmod;
    const int rem = row % ROWS_PER_SIDE;
    const int t = rem / B_, b = rem % B_;
    const int tok = x[b * (2 * T_) + side * T_ + t];
    const float* arow = emb + (size_t)tok * D_;

    const int ncol = ntile * 16 + lmod;
    const float* brow = Wih + (size_t)ncol * D_;       // B[k][n] = Wih[n][k]

    v8f acc = {};
    #pragma unroll 5
    for (int k = 0; k < D_; k += 4) {
        v2f a, bb;
        a.x  = arow[k + kof];  a.y  = arow[k + kof + 1];
        bb.x = brow[k + kof];  bb.y = brow[k + kof + 1];
        acc = wmma_f32(a, bb, acc);
    }
    const float bias = bih[ncol];
    const int rbase = mtile * 16 + (lane >> 4) * 8;    // C/D layout: m = v + 8*half
    #pragma unroll
    for (int v = 0; v < 8; ++v)
        xp[(size_t)(rbase + v) * G_ + ncol] = acc[v] + bias;
}

// Kernel 3: input projection, layer 1: xp[row,g] = h0[row,:] @ Wih1[g,:] + bih1
__global__ void k_xp1(const float* __restrict__ h0,
                      const float* __restrict__ WihC, const float* __restrict__ bihC,
                      const float* __restrict__ WihE, const float* __restrict__ bihE,
                      float* __restrict__ xp)
{
    const int tid  = threadIdx.x;
    const int lane = tid & 31, wave = tid >> 5;
    const int gw   = blockIdx.x * 8 + wave;
    const int mtile = gw / 12, ntile = gw % 12;
    const int lmod = lane & 15;
    const int kof  = (lane >> 4) * 2;
    const int side = (mtile * 16) / ROWS_PER_SIDE;
    const float* Wih = side ? WihE : WihC;
    const float* bih = side ? bihE : bihC;

    const int row = mtile * 16 + lmod;
    const float* arow = h0 + (size_t)row * H_;
    const int ncol = ntile * 16 + lmod;
    const float* brow = Wih + (size_t)ncol * H_;

    v8f acc = {};
    #pragma unroll
    for (int k = 0; k < H_; k += 4) {
        v2f a, bb;
        a.x  = arow[k + kof];  a.y  = arow[k + kof + 1];
        bb.x = brow[k + kof];  bb.y = brow[k + kof + 1];
        acc = wmma_f32(a, bb, acc);
    }
    const float bias = bih[ncol];
    const int rbase = mtile * 16 + (lane >> 4) * 8;
    #pragma unroll
    for (int v = 0; v < 8; ++v)
        xp[(size_t)(rbase + v) * G_ + ncol] = acc[v] + bias;
}

// ---------------------------------------------------------------------------
// Kernel 2/4: GRU recurrence, one layer, both sides (blockIdx.x = side).
// 16 waves; wave (mtile=w&3, jg=w>>2) owns the r/z/n 16x16 tiles for the same
// (batch rows, j columns): gate math stays on WMMA accumulators in registers.
// Whh B-fragments are hoisted into registers for all 1000 steps (48 v2f/wave);
// xp gate inputs are issued BEFORE the WMMA loop so global latency hides
// under the 48 WMMAs; h ping-pongs in LDS with one barrier per step.
// ---------------------------------------------------------------------------
__global__ void __launch_bounds__(512)
k_recur(const float* __restrict__ xp, float* __restrict__ hout,
        const float* __restrict__ WhhC, const float* __restrict__ bhhC,
        const float* __restrict__ WhhE, const float* __restrict__ bhhE)
{
    __shared__ float hbuf[2][B_ * H_];                 // 2 x 16 KB ping-pong

    const int side = blockIdx.x;
    const float* Whh = side ? WhhE : WhhC;             // [192][64] row-major
    const float* bhh = side ? bhhE : bhhC;

    const int tid  = threadIdx.x;
    const int lane = tid & 31, wave = tid >> 5;        // 16 waves
    const int lmod = lane & 15;
    const int kof  = (lane >> 4) * 2;
    const int rbo  = (lane >> 4) * 8;
    const int mtile = wave & 3;                        // batch tile
    const int jg    = wave >> 2;                       // j-column group (0..3)
    const int m_base = mtile * 16;
    const int jcol   = jg * 16 + lmod;                 // j in 0..63

    for (int i = tid; i < B_ * H_; i += blockDim.x) hbuf[0][i] = 0.0f;
    __syncthreads();

    const float bh_r = bhh[jcol];
    const float bh_z = bhh[H_ + jcol];
    const float bh_n = bhh[2 * H_ + jcol];

    // Hoist Whh B-fragments (B[k][n] = Whh[n][k]) into registers for all t.
    const float* wr = Whh + (size_t)(0 * H_ + jcol) * H_;
    const float* wz = Whh + (size_t)(1 * H_ + jcol) * H_;
    const float* wn = Whh + (size_t)(2 * H_ + jcol) * H_;
    v2f Br[16], Bz[16], Bn[16];
    #pragma unroll
    for (int kk = 0; kk < 16; ++kk) {
        const int k = kk * 4 + kof;
        Br[kk].x = wr[k]; Br[kk].y = wr[k + 1];
        Bz[kk].x = wz[k]; Bz[kk].y = wz[k + 1];
        Bn[kk].x = wn[k]; Bn[kk].y = wn[k + 1];
    }

    const float* xp_s = xp   + (size_t)side * T_ * B_ * G_;
    float*       ho_s = hout + (size_t)side * T_ * B_ * H_;

    int cur = 0;
    for (int t = 0; t < T_; ++t) {
        const float* xslab = xp_s + (size_t)t * B_ * G_;   // contiguous 48 KB
        // Preload this step's gate inputs (independent of h -> hides under WMMA)
        float xr[8], xz[8], xn[8];
        #pragma unroll
        for (int v = 0; v < 8; ++v) {
            const float* p = xslab + (m_base + v + rbo) * G_;
            xr[v] = p[jcol];
            xz[v] = p[H_ + jcol];
            xn[v] = p[2 * H_ + jcol];
        }
        // one-prefetch-per-thread covers the next 48 KB slab (512 * 96 B)
        if (t + 1 < T_) __builtin_prefetch(xslab + B_ * G_ + tid * 24, 0, 1);

        const float* h = hbuf[cur];
        v8f ar = {}, az = {}, an = {};
        #pragma unroll
        for (int kk = 0; kk < 16; ++kk) {
            const int k = kk * 4 + kof;
            v2f a;
            a.x = h[(m_base + lmod) * H_ + k];
            a.y = h[(m_base + lmod) * H_ + k + 1];
            ar = wmma_f32(a, Br[kk], ar);
            az = wmma_f32(a, Bz[kk], az);
            an = wmma_f32(a, Bn[kk], an);
        }
        float* hn_buf = hbuf[cur ^ 1];
        float* hoslab = ho_s + (size_t)t * B_ * H_;        // contiguous 16 KB
        #pragma unroll
        for (int v = 0; v < 8; ++v) {
            const int brow = m_base + v + rbo;             // batch index 0..63
            const float r  = sigmoid_(xr[v] + ar[v] + bh_r);
            const float z  = sigmoid_(xz[v] + az[v] + bh_z);
            const float nn = tanhf(xn[v] + r * (an[v] + bh_n));
            const float hold = h[brow * H_ + jcol];
            const float hnew = (1.0f - z) * nn + z * hold;
            hn_buf[brow * H_ + jcol] = hnew;
            hoslab[brow * H_ + jcol] = hnew;
        }
        __syncthreads();
        cur ^= 1;
    }
}

// ---------------------------------------------------------------------------
// Kernel 5: FC head. One wave handles a 16-token tile through fc1..fc3 with
// WMMA (per-wave LDS activation scratch), then fc4+softmax on VALU.
// Token rows are t-major (row = t*64+b); output is [b][t][2].
// ---------------------------------------------------------------------------
__global__ void __launch_bounds__(128)
k_fc(const float* __restrict__ h1,   // [2][1000][64][64] t-major
     const float* __restrict__ fc1w, const float* __restrict__ fc1b,
     const float* __restrict__ fc2w, const float* __restrict__ fc2b,
     const float* __restrict__ fc3w, const float* __restrict__ fc3b,
     const float* __restrict__ fc4w, const float* __restrict__ fc4b,
     float* __restrict__ out)
{
    __shared__ float buf1[4][16 * 128];   // per-wave act1 (reused for act3)
    __shared__ float buf2[4][16 * 64];    // per-wave act2

    const int tid  = threadIdx.x;
    const int lane = tid & 31, wave = tid >> 5;        // 4 waves
    const int lmod = lane & 15;
    const int kof  = (lane >> 4) * 2;
    const int rbo  = (lane >> 4) * 8;
    const int mtile = blockIdx.x * 4 + wave;           // 0..3999
    const int row0  = mtile * 16;                      // t-major token rows
    const int arow  = row0 + lmod;
    float* A1 = buf1[wave];
    float* A2 = buf2[wave];

    // fc1: in=128 (caller||callee), out=128, relu
    for (int nt = 0; nt < 8; ++nt) {
        const int ncol = nt * 16 + lmod;
        v8f acc = {};
        #pragma unroll
        for (int k = 0; k < 128; k += 4) {
            const float* src = (k < 64)
                ? (h1 + (size_t)arow * H_ + k)
                : (h1 + (size_t)(ROWS_PER_SIDE + arow) * H_ + (k - 64));
            v2f a, bb;
            a.x  = src[kof];                     a.y  = src[kof + 1];
            bb.x = fc1w[ncol * 128 + k + kof];   bb.y = fc1w[ncol * 128 + k + kof + 1];
            acc = wmma_f32(a, bb, acc);
        }
        const float bias = fc1b[ncol];
        #pragma unroll
        for (int v = 0; v < 8; ++v) {
            float val = acc[v] + bias;
            A1[(v + rbo) * 128 + ncol] = val > 0.0f ? val : 0.0f;
        }
    }
    // fc2: in=128, out=64, relu
    for (int nt = 0; nt < 4; ++nt) {
        const int ncol = nt * 16 + lmod;
        v8f acc = {};
        #pragma unroll
        for (int k = 0; k < 128; k += 4) {
            v2f a, bb;
            a.x  = A1[lmod * 128 + k + kof];     a.y  = A1[lmod * 128 + k + kof + 1];
            bb.x = fc2w[ncol * 128 + k + kof];   bb.y = fc2w[ncol * 128 + k + kof + 1];
            acc = wmma_f32(a, bb, acc);
        }
        const float bias = fc2b[ncol];
        #pragma unroll
        for (int v = 0; v < 8; ++v) {
            float val = acc[v] + bias;
            A2[(v + rbo) * 64 + ncol] = val > 0.0f ? val : 0.0f;
        }
    }
    // fc3: in=64, out=32, relu -> A1 reused as [16][32]
    for (int nt = 0; nt < 2; ++nt) {
        const int ncol = nt * 16 + lmod;
        v8f acc = {};
        #pragma unroll
        for (int k = 0; k < 64; k += 4) {
            v2f a, bb;
            a.x  = A2[lmod * 64 + k + kof];      a.y  = A2[lmod * 64 + k + kof + 1];
            bb.x = fc3w[ncol * 64 + k + kof];    bb.y = fc3w[ncol * 64 + k + kof + 1];
            acc = wmma_f32(a, bb, acc);
        }
        const float bias = fc3b[ncol];
        #pragma unroll
        for (int v = 0; v < 8; ++v) {
            float val = acc[v] + bias;
            A1[(v + rbo) * 32 + ncol] = val > 0.0f ? val : 0.0f;
        }
    }
    // fc4 (2 outputs) + softmax: lanes 0..15, one token each
    if (lane < 16) {
        const int m = lmod;
        float l0 = fc4b[0], l1 = fc4b[1];
        #pragma unroll
        for (int k = 0; k < 32; ++k) {
            const float v = A1[m * 32 + k];
            l0 += v * fc4w[k];
            l1 += v * fc4w[32 + k];
        }
        const float mx = fmaxf(l0, l1);
        const float e0 = __expf(l0 - mx), e1 = __expf(l1 - mx);
        const float inv = 1.0f / (e0 + e1);
        const int r = row0 + m;                 // t-major row
        const int b = r % B_, t = r / B_;
        out[(size_t)(b * T_ + t) * 2 + 0] = e0 * inv;
        out[(size_t)(b * T_ + t) * 2 + 1] = e1 * inv;
    }
}

// ---------------------------------------------------------------------------
extern "C" void kernel_launch(void* const* d_in, const int* in_sizes, int n_in,
                              void* d_out, int out_size, void* d_ws, size_t ws_size,
                              hipStream_t stream)
{
    const int*   x     = (const int*)  d_in[0];
    const float* emb   = (const float*)d_in[1];
    const float* cWih0 = (const float*)d_in[2];
    const float* cWhh0 = (const float*)d_in[3];
    const float* cbih0 = (const float*)d_in[4];
    const float* cbhh0 = (const float*)d_in[5];
    const float* cWih1 = (const float*)d_in[6];
    const float* cWhh1 = (const float*)d_in[7];
    const float* cbih1 = (const float*)d_in[8];
    const float* cbhh1 = (const float*)d_in[9];
    const float* eWih0 = (const float*)d_in[10];
    const float* eWhh0 = (const float*)d_in[11];
    const float* ebih0 = (const float*)d_in[12];
    const float* ebhh0 = (const float*)d_in[13];
    const float* eWih1 = (const float*)d_in[14];
    const float* eWhh1 = (const float*)d_in[15];
    const float* ebih1 = (const float*)d_in[16];
    const float* ebhh1 = (const float*)d_in[17];
    const float* fc1w  = (const float*)d_in[18];
    const float* fc1b  = (const float*)d_in[19];
    const float* fc2w  = (const float*)d_in[20];
    const float* fc2b  = (const float*)d_in[21];
    const float* fc3w  = (const float*)d_in[22];
    const float* fc3b  = (const float*)d_in[23];
    const float* fc4w  = (const float*)d_in[24];
    const float* fc4b  = (const float*)d_in[25];
    float* out = (float*)d_out;

    float* ws = (float*)d_ws;
    float* xp = ws;                                        // 2*64000*192 floats (reused L0/L1)
    float* h0 = ws + (size_t)ROWS_TOTAL * G_;              // 2*64000*64
    float* h1 = h0 + (size_t)ROWS_TOTAL * H_;              // 2*64000*64

    // 8000 M-tiles * 12 N-tiles = 96000 waves / 8 waves per block
    k_xp0 <<<12000, 256, 0, stream>>>(x, emb, cWih0, cbih0, eWih0, ebih0, xp);
    k_recur<<<2,    512, 0, stream>>>(xp, h0, cWhh0, cbhh0, eWhh0, ebhh0);
    k_xp1 <<<12000, 256, 0, stream>>>(h0, cWih1, cbih1, eWih1, ebih1, xp);
    k_recur<<<2,    512, 0, stream>>>(xp, h1, cWhh1, cbhh1, eWhh1, ebhh1);
    k_fc  <<<1000,  128, 0, stream>>>(h1, fc1w, fc1b, fc2w, fc2b,
                                      fc3w, fc3b, fc4w, fc4b, out);
}